// ConvertParamMomentsTEtoParamsTE_16277926052108
// MI455X (gfx1250) — compile-verified
//
#include <hip/hip_runtime.h>

typedef __attribute__((ext_vector_type(2))) float v2f;
typedef __attribute__((ext_vector_type(8))) float v8f;

#define NH 32
#define NV 64
#define WAVES 4
#define CW_STRIDE 72   // 2*72 % 64 = 16 -> half-lane groups hit disjoint banks
#define CH_STRIDE 36   // 36*4 = 144 bytes, 16B-aligned rows for float4 LDS stores
#define WAVE_LDS (NH*CW_STRIDE + NH*CH_STRIDE + 3*NH) // 2304 + 1152 + 96 = 3552 floats

__global__ __launch_bounds__(WAVES * 32)
void cpm_te_kernel(const float* __restrict__ muvTE,
                   const float* __restrict__ covvhTE,
                   const float* __restrict__ covhTE,
                   const float* __restrict__ covh_diag,
                   const float* __restrict__ muh,
                   const float* __restrict__ covvh,
                   const float* __restrict__ muhTE,
                   const float* __restrict__ covvbarTE,
                   float* __restrict__ out,
                   int Btot)
{
    __shared__ __align__(16) float smem[WAVES * WAVE_LDS];
    const int lane = threadIdx.x & 31;
    const int wave = threadIdx.x >> 5;
    const int b = blockIdx.x * WAVES + wave;
    if (b >= Btot) return;   // uniform per wave -> EXEC stays all-ones inside

    float* cw     = smem + wave * WAVE_LDS;     // Cw = cinv ⊙ covvh  [32 x 64], stride 72
    float* ch     = cw + NH * CW_STRIDE;        // covhTE            [32 x 32], stride 36
    float* cinv_s = ch + NH * CH_STRIDE;        // [32]
    float* v1_s   = cinv_s + NH;                // [32]
    float* g_s    = v1_s + NH;                  // [32]

    const float* covhTE_b  = covhTE  + (size_t)b * (NH * NH);
    const float* covvhTE_b = covvhTE + (size_t)b * (NH * NV);
    const float* covvh_b   = covvh   + (size_t)b * (NH * NV);
    const float* muvTE_b   = muvTE   + (size_t)b * NV;

    const size_t Bsz = (size_t)Btot;
    float* out_b  = out;                          // bTE    [B, NV]
    float* out_wt = out + Bsz * NV;               // wtTE   [B, NH, NV]
    float* out_sg = out + Bsz * (NV + NH * NV);   // sig2TE [B]
    float* out_mh = out_sg + Bsz;                 // muhTE  [B, NH]
    float* out_ch = out_mh + Bsz * NH;            // covhTE [B, NH, NH]

    // ---- per-lane h scalars -------------------------------------------------
    const float cinv = 1.0f / covh_diag[(size_t)b * NH + lane];
    const float v1   = cinv * muh[(size_t)b * NH + lane];
    cinv_s[lane] = cinv;
    v1_s[lane]   = v1;

    // ---- stage covhTE into LDS (stride 36) + passthrough copy to output ----
#pragma unroll
    for (int j = 0; j < 8; ++j) {
        const int gi = (j * 32 + lane) * 4;          // 0..1023, float4 granules
        const float4 val = *(const float4*)(covhTE_b + gi);
        *(float4*)(out_ch + (size_t)b * (NH * NH) + gi) = val;
        const int row = gi >> 5, col = gi & 31;
        *(float4*)(ch + row * CH_STRIDE + col) = val;
    }

    // ---- stage Cw = cinv ⊙ covvh into LDS (stride 72) -----------------------
#pragma unroll
    for (int j = 0; j < 16; ++j) {
        const int gi = (j * 32 + lane) * 4;          // 0..2047
        float4 val = *(const float4*)(covvh_b + gi);
        const int h = gi >> 6, col = gi & 63;
        const float s = cinv_s[h];
        val.x *= s; val.y *= s; val.z *= s; val.w *= s;
        *(float4*)(cw + h * CW_STRIDE + col) = val;
    }

    // ---- t = covhTE @ v1 ; g = t - muhTE ; muhTE passthrough ----------------
    float t = 0.0f;
#pragma unroll
    for (int k = 0; k < NH; ++k)
        t += ch[lane * CH_STRIDE + k] * v1_s[k];
    const float mhte = muhTE[(size_t)b * NH + lane];
    out_mh[(size_t)b * NH + lane] = mhte;
    g_s[lane] = t - mhte;

    // ---- batched GEMM M = covhTE @ Cw via V_WMMA_F32_16X16X4_F32 ------------
    // Fused epilogue: wtTE = cinv*(covvhTE - M), tr += Cw*(2*covvhTE - M)
    const int m_a = lane & 15;        // A: row within tile
    const int khi = lane >> 4;        // half-lane group
    const int n_c = lane & 15;        // B/C/D: column within tile
    float tr = 0.0f;

#pragma unroll
    for (int Mi = 0; Mi < 2; ++Mi) {
#pragma unroll
        for (int Nj = 0; Nj < 4; ++Nj) {
            v8f acc = {};
#pragma unroll
            for (int kk = 0; kk < 8; ++kk) {
                v2f a, bm;
                // A 16x4 tile: VGPR0/1 = K pair (2*khi, 2*khi+1)
                const int arow = (16 * Mi + m_a) * CH_STRIDE + 4 * kk + 2 * khi;
                a.x = ch[arow];
                a.y = ch[arow + 1];
                // B 4x16 tile: VGPR0/1 = rows kb, kb+1 (kb = 2*khi)
                const int kb = 2 * khi;
                bm.x = cw[(4 * kk + kb)     * CW_STRIDE + 16 * Nj + n_c];
                bm.y = cw[(4 * kk + kb + 1) * CW_STRIDE + 16 * Nj + n_c];
                acc = __builtin_amdgcn_wmma_f32_16x16x4_f32(
                        false, a, false, bm, (short)0, acc, false, false);
            }
#pragma unroll
            for (int r = 0; r < 8; ++r) {
                const int h = 16 * Mi + r + 8 * khi;   // C/D row mapping
                const int v = 16 * Nj + n_c;
                const float cvte = covvhTE_b[h * NV + v];
                const float mval = acc[r];
                out_wt[(size_t)b * (NH * NV) + h * NV + v] =
                    cinv_s[h] * (cvte - mval);
                tr += cw[h * CW_STRIDE + v] * (2.0f * cvte - mval);
            }
        }
    }

    // ---- bTE[v] = muvTE[v] - covvhTE^T v1 + Cw^T g --------------------------
    float b0 = muvTE_b[lane];
    float b1 = muvTE_b[lane + 32];
#pragma unroll 4
    for (int h = 0; h < NH; ++h) {
        const float vv1 = v1_s[h], gg = g_s[h];
        b0 += cw[h * CW_STRIDE + lane]      * gg - covvhTE_b[h * NV + lane]      * vv1;
        b1 += cw[h * CW_STRIDE + lane + 32] * gg - covvhTE_b[h * NV + lane + 32] * vv1;
    }
    out_b[(size_t)b * NV + lane]      = b0;
    out_b[(size_t)b * NV + lane + 32] = b1;

    // ---- sig2TE = (covvbarTE - tr) / NV -------------------------------------
#pragma unroll
    for (int off = 16; off > 0; off >>= 1)
        tr += __shfl_xor(tr, off, 32);
    if (lane == 0)
        out_sg[b] = (covvbarTE[b] - tr) * (1.0f / (float)NV);
}

extern "C" void kernel_launch(void* const* d_in, const int* in_sizes, int n_in,
                              void* d_out, int out_size, void* d_ws, size_t ws_size,
                              hipStream_t stream) {
    const float* muvTE     = (const float*)d_in[0];
    const float* covvhTE   = (const float*)d_in[1];
    const float* covhTE    = (const float*)d_in[2];
    const float* covh_diag = (const float*)d_in[3];
    const float* muh       = (const float*)d_in[4];
    const float* covvh     = (const float*)d_in[5];
    const float* muhTE     = (const float*)d_in[6];
    const float* covvbarTE = (const float*)d_in[7];
    const int B = in_sizes[7];   // covvbarTE is [B]

    const int grid = (B + WAVES - 1) / WAVES;
    cpm_te_kernel<<<grid, WAVES * 32, 0, stream>>>(
        muvTE, covvhTE, covhTE, covh_diag, muh, covvh, muhTE, covvbarTE,
        (float*)d_out, B);
}